// GCN_1649267442174
// MI455X (gfx1250) — compile-verified
//
#include <hip/hip_runtime.h>

#define N_NODES 50000
#define N_GENES 1000
#define N_EDGES 1600000
#define HIDDEN  64

typedef __attribute__((ext_vector_type(2))) float v2f;
typedef __attribute__((ext_vector_type(8))) float v8f;

// D = A(16x4 f32) * B(4x16 f32) + C(16x16 f32), full fp32 on the matrix pipe.
// 8-arg pattern: (neg_a, A, neg_b, B, c_mod, C, reuse_a, reuse_b)
__device__ __forceinline__ v8f wmma4(v2f a, v2f b, v8f c) {
  return __builtin_amdgcn_wmma_f32_16x16x4_f32(false, a, false, b, (short)0, c,
                                               false, false);
}

__global__ void k_zero4(float4* __restrict__ p, int n4) {
  int i = blockIdx.x * blockDim.x + threadIdx.x;
  if (i < n4) p[i] = make_float4(0.f, 0.f, 0.f, 0.f);
}

__global__ void k_degree(const int* __restrict__ src, const int* __restrict__ dst,
                         float* __restrict__ deg_out, float* __restrict__ deg_in,
                         int ne) {
  int e = blockIdx.x * blockDim.x + threadIdx.x;
  if (e < ne) {
    atomicAdd(&deg_out[src[e]], 1.0f);
    atomicAdd(&deg_in[dst[e]], 1.0f);
  }
}

__global__ void k_norm(const float* __restrict__ deg_out,
                       const float* __restrict__ deg_in,
                       float* __restrict__ nsrc, float* __restrict__ ndst, int n) {
  int i = blockIdx.x * blockDim.x + threadIdx.x;
  if (i < n) {
    nsrc[i] = rsqrtf(fmaxf(deg_out[i], 1.0f));
    ndst[i] = rsqrtf(fmaxf(deg_in[i], 1.0f));
  }
}

// hs0[i,j] = (sum_k [x_u|x_s][i,k] * W0[k,j]) * norm_src[i]
// One wave per 16-row M tile; 4 N-tiles (64 cols) accumulated in registers.
__global__ void __launch_bounds__(128)
k_gemm1(const float* __restrict__ xu, const float* __restrict__ xs,
        const float* __restrict__ W0, const float* __restrict__ nsrc,
        float* __restrict__ hs0) {
  const int lane = threadIdx.x & 31;
  const int wave = (blockIdx.x * blockDim.x + threadIdx.x) >> 5;
  const int MT = N_NODES / 16; // 3125
  if (wave >= MT) return;      // wave-uniform: EXEC stays all-ones for WMMA
  const int m0  = wave * 16;
  const int r   = lane & 15;          // A: row-in-tile, B/D: col-in-tile
  const int kk  = (lane >> 4) * 2;    // K sub-offset (0 or 2)
  const int row = m0 + r;

  const v8f vz = {0.f, 0.f, 0.f, 0.f, 0.f, 0.f, 0.f, 0.f};
  v8f acc[4] = {vz, vz, vz, vz};

  // K = 0..999 : x_u against W0 rows [0,1000)
  for (int k0 = 0; k0 < N_GENES; k0 += 4) {
    v2f a;
    a.x = xu[row * N_GENES + k0 + kk];
    a.y = xu[row * N_GENES + k0 + kk + 1];
#pragma unroll
    for (int j = 0; j < 4; ++j) {
      const int col = j * 16 + r;
      v2f b;
      b.x = W0[(k0 + kk) * HIDDEN + col];
      b.y = W0[(k0 + kk + 1) * HIDDEN + col];
      acc[j] = wmma4(a, b, acc[j]);
    }
  }
  // K = 1000..1999 : x_s against W0 rows [1000,2000)
  for (int k0 = 0; k0 < N_GENES; k0 += 4) {
    v2f a;
    a.x = xs[row * N_GENES + k0 + kk];
    a.y = xs[row * N_GENES + k0 + kk + 1];
#pragma unroll
    for (int j = 0; j < 4; ++j) {
      const int col = j * 16 + r;
      v2f b;
      b.x = W0[(N_GENES + k0 + kk) * HIDDEN + col];
      b.y = W0[(N_GENES + k0 + kk + 1) * HIDDEN + col];
      acc[j] = wmma4(a, b, acc[j]);
    }
  }

  // D layout: lane l, vgpr v -> (M=(l>>4)*8+v, N=l&15). Fold in norm_src.
#pragma unroll
  for (int v = 0; v < 8; ++v) {
    const int orow = m0 + ((lane >> 4) * 8) + v;
    const float ns = nsrc[orow];
#pragma unroll
    for (int j = 0; j < 4; ++j)
      hs0[orow * HIDDEN + j * 16 + r] = acc[j][v] * ns;
  }
}

// agg[dst[e], :] += hin[src[e], :]. One thread per (edge, 16B chunk):
// float4 gather (global_load_b128, coalesced across the 256B row) + 4 f32 atomics.
// Feature maps are 12.8 MB each -> gathers and atomics resolve in L2, not HBM.
__global__ void k_scatter(const int* __restrict__ src, const int* __restrict__ dst,
                          const float* __restrict__ hin, float* __restrict__ agg) {
  int t = blockIdx.x * blockDim.x + threadIdx.x;
  int e = t >> 4;          // edge
  int q = (t & 15) << 2;   // feature offset 0,4,...,60
  if (e < N_EDGES) {
    const float4 v = *(const float4*)(hin + src[e] * HIDDEN + q);
    float* a = agg + dst[e] * HIDDEN + q;
    atomicAdd(a + 0, v.x);
    atomicAdd(a + 1, v.y);
    atomicAdd(a + 2, v.z);
    atomicAdd(a + 3, v.w);
  }
}

// hs1 = relu(agg0 * norm_dst + b0) * norm_src  (fused layer-0 epilogue + layer-1 prologue)
// Vectorized: one thread per 4 features.
__global__ void k_post0(const float4* __restrict__ agg0, const float* __restrict__ ndst,
                        const float* __restrict__ nsrc, const float4* __restrict__ b0,
                        float4* __restrict__ hs1) {
  int i = blockIdx.x * blockDim.x + threadIdx.x;      // chunk index
  if (i < N_NODES * (HIDDEN / 4)) {
    const int node = i >> 4;                          // 16 chunks per node
    const float nd = ndst[node], ns = nsrc[node];
    const float4 a = agg0[i];
    const float4 b = b0[i & 15];
    float4 o;
    o.x = fmaxf(a.x * nd + b.x, 0.f) * ns;
    o.y = fmaxf(a.y * nd + b.y, 0.f) * ns;
    o.z = fmaxf(a.z * nd + b.z, 0.f) * ns;
    o.w = fmaxf(a.w * nd + b.w, 0.f) * ns;
    hs1[i] = o;
  }
}

// Per wave: one 16-row tile x one 16-gene column tile. Computes BOTH the beta tile
// (x cols c0..) and the gamma tile (x cols 1000+c0..) so the final
// out = beta*x_u + gamma*x_s fuses into the epilogue (no 400MB intermediate).
__global__ void __launch_bounds__(128)
k_gemm2(const float* __restrict__ agg1, const float* __restrict__ ndst,
        const float* __restrict__ W1, const float* __restrict__ b1,
        const float* __restrict__ xu, const float* __restrict__ xs,
        float* __restrict__ out) {
  const int lane = threadIdx.x & 31;
  const int wave = (blockIdx.x * blockDim.x + threadIdx.x) >> 5;
  const int MT = N_NODES / 16;          // 3125
  const int CT = (N_GENES + 15) / 16;   // 63 (last tile ragged: 1000 = 62*16+8)
  if (wave >= MT * CT) return;          // wave-uniform guard
  const int mt = wave % MT;
  const int ct = wave / MT;
  const int m0 = mt * 16;
  const int c0 = ct * 16;
  const int r  = lane & 15;
  const int kk = (lane >> 4) * 2;
  const int row = m0 + r;
  const float nd = ndst[row];           // fold norm_dst into A load

  const int W1LD  = 2 * N_GENES;        // 2000
  const int colB  = c0 + r;             // beta column in x, <= 1007 < 2000: safe
  const int colG  = N_GENES + c0 + r;   // gamma column in x
  const int colGc = (colG < W1LD) ? colG : (W1LD - 1); // clamp ragged B loads

  v8f accB = {0.f, 0.f, 0.f, 0.f, 0.f, 0.f, 0.f, 0.f};
  v8f accG = {0.f, 0.f, 0.f, 0.f, 0.f, 0.f, 0.f, 0.f};

#pragma unroll
  for (int k0 = 0; k0 < HIDDEN; k0 += 4) {
    v2f a;
    a.x = agg1[row * HIDDEN + k0 + kk] * nd;
    a.y = agg1[row * HIDDEN + k0 + kk + 1] * nd;
    v2f bB, bG;
    bB.x = W1[(k0 + kk) * W1LD + colB];
    bB.y = W1[(k0 + kk + 1) * W1LD + colB];
    bG.x = W1[(k0 + kk) * W1LD + colGc];
    bG.y = W1[(k0 + kk + 1) * W1LD + colGc];
    accB = wmma4(a, bB, accB);
    accG = wmma4(a, bG, accG);
  }

  const float biasB = b1[colB];
  const float biasG = b1[colGc];
  const int c = c0 + r; // gene index of this lane's D column
#pragma unroll
  for (int v = 0; v < 8; ++v) {
    const int orow = m0 + ((lane >> 4) * 8) + v;
    if (c < N_GENES) { // mask only the ragged stores; WMMA already retired
      const float beta  = accB[v] + biasB;
      const float gamma = accG[v] + biasG;
      const int idx = orow * N_GENES + c;
      out[idx] = beta * xu[idx] + gamma * xs[idx];
    }
  }
}

extern "C" void kernel_launch(void* const* d_in, const int* in_sizes, int n_in,
                              void* d_out, int out_size, void* d_ws, size_t ws_size,
                              hipStream_t stream) {
  const float* xu = (const float*)d_in[0];
  const float* xs = (const float*)d_in[1];
  const float* W0 = (const float*)d_in[2];
  const float* b0 = (const float*)d_in[3];
  const float* W1 = (const float*)d_in[4];
  const float* b1 = (const float*)d_in[5];
  const int*  src = (const int*)d_in[6];
  const int*  dst = (const int*)d_in[7];
  float*      out = (float*)d_out;

  // Workspace layout (floats). Accumulated buffers first so one zero-fill covers them.
  float* ws      = (float*)d_ws;
  float* deg_out = ws;                       //   50,000
  float* deg_in  = deg_out + N_NODES;        //   50,000
  float* agg0    = deg_in + N_NODES;         // 3,200,000
  float* agg1    = agg0 + N_NODES * HIDDEN;  // 3,200,000
  float* nsrc    = agg1 + N_NODES * HIDDEN;  //   50,000
  float* ndst    = nsrc + N_NODES;           //   50,000
  float* hs0     = ndst + N_NODES;           // 3,200,000
  float* hs1     = hs0 + N_NODES * HIDDEN;   // 3,200,000  (total 52 MB)

  const int ZN4 = (2 * N_NODES + 2 * N_NODES * HIDDEN) / 4; // 6.5M floats / 4
  k_zero4<<<(ZN4 + 255) / 256, 256, 0, stream>>>((float4*)ws, ZN4);

  k_degree<<<(N_EDGES + 255) / 256, 256, 0, stream>>>(src, dst, deg_out, deg_in, N_EDGES);
  k_norm<<<(N_NODES + 255) / 256, 256, 0, stream>>>(deg_out, deg_in, nsrc, ndst, N_NODES);

  const int MT = N_NODES / 16;                       // 3125 waves
  k_gemm1<<<(MT + 3) / 4, 128, 0, stream>>>(xu, xs, W0, nsrc, hs0);

  k_scatter<<<(N_EDGES * 16) / 256, 256, 0, stream>>>(src, dst, hs0, agg0);
  k_post0<<<(N_NODES * (HIDDEN / 4) + 255) / 256, 256, 0, stream>>>(
      (const float4*)agg0, ndst, nsrc, (const float4*)b0, (float4*)hs1);
  k_scatter<<<(N_EDGES * 16) / 256, 256, 0, stream>>>(src, dst, hs1, agg1);

  const int W2 = MT * ((N_GENES + 15) / 16);         // 196,875 wave-tasks
  k_gemm2<<<(W2 + 3) / 4, 128, 0, stream>>>(agg1, ndst, W1, b1, xu, xs, out);
}